// SimpleGATLayer_67138928771585
// MI455X (gfx1250) — compile-verified
//
#include <hip/hip_runtime.h>
#include <hip/hip_bf16.h>

typedef float v2f __attribute__((ext_vector_type(2)));
typedef float v8f __attribute__((ext_vector_type(8)));

#define IN_DIM 128
#define OUT_DIM 64

// Force the native global_atomic_add_f32 path (no CAS expansion).
__device__ __forceinline__ void atomic_add_f32(float* p, float v) {
  unsafeAtomicAdd(p, v);
}

// ---------------------------------------------------------------------------
// Kernel 1: h = x @ W^T using V_WMMA_F32_16X16X4_F32.
// Block = 128 threads = 4 waves. Each block handles 16 node rows; wave w
// produces columns [16w, 16w+16). K loop: 128 in steps of 4.
//
// A (16x4 f32) lane layout (ISA 7.12.2): lanes 0-15 hold M=lane, VGPR0=K0,
// VGPR1=K1; lanes 16-31 hold M=lane-16, VGPR0=K2, VGPR1=K3.
// B (4x16 f32): lanes 0-15 hold N=lane with K0/K1, lanes 16-31 K2/K3.
// Both therefore load (row = lane&15, k = kk + 2*(lane>>4)) pairs.
// D (16x16 f32): VGPR r holds M = r + 8*(lane>>4), N = lane&15.
// ---------------------------------------------------------------------------
__global__ __launch_bounds__(128) void gat_gemm_kernel(
    const float* __restrict__ x, const float* __restrict__ W,
    float* __restrict__ h, int N) {
  const int wave = threadIdx.x >> 5;   // 0..3 -> output column tile
  const int lane = threadIdx.x & 31;
  const int l = lane & 15;
  const int g = lane >> 4;

  const long mrow = (long)blockIdx.x * 16 + l;          // node row for A
  const float* __restrict__ arow = x + mrow * IN_DIM + 2 * g;
  const float* __restrict__ brow = W + (long)(wave * 16 + l) * IN_DIM + 2 * g;

  v8f c = {};
#pragma unroll
  for (int kk = 0; kk < IN_DIM; kk += 4) {
    v2f a = *(const v2f*)(arow + kk);  // x[row][kk+2g], x[row][kk+2g+1]
    v2f b = *(const v2f*)(brow + kk);  // W[n][kk+2g],  W[n][kk+2g+1]
    c = __builtin_amdgcn_wmma_f32_16x16x4_f32(
        /*neg_a=*/false, a, /*neg_b=*/false, b,
        /*c_mod=*/(short)0, c, /*reuse_a=*/false, /*reuse_b=*/false);
  }

  // Store D tile: h[(tile*16 + r + 8g)][wave*16 + l]
  long base = ((long)blockIdx.x * 16 + 8 * g) * OUT_DIM + wave * 16 + l;
#pragma unroll
  for (int r = 0; r < 8; ++r) {
    h[base + (long)r * OUT_DIM] = c[r];
  }
}

// ---------------------------------------------------------------------------
// Kernel 2: per-node attention logits  s_src[n] = h[n,:]·a_src,
// s_dst[n] = h[n,:]·a_dst.  One wave per node, coalesced 256B row loads,
// butterfly shuffle reduction (wave32).
// ---------------------------------------------------------------------------
__global__ __launch_bounds__(256) void gat_logits_kernel(
    const float* __restrict__ h, const float* __restrict__ a_src,
    const float* __restrict__ a_dst, float* __restrict__ s_src,
    float* __restrict__ s_dst, int N) {
  const int node = (int)((blockIdx.x * (long)blockDim.x + threadIdx.x) >> 5);
  const int lane = threadIdx.x & 31;
  if (node >= N) return;

  const float* __restrict__ hr = h + (long)node * OUT_DIM;
  float v0 = hr[lane];
  float v1 = hr[lane + 32];
  float ps = v0 * a_src[lane] + v1 * a_src[lane + 32];
  float pd = v0 * a_dst[lane] + v1 * a_dst[lane + 32];

#pragma unroll
  for (int off = 16; off > 0; off >>= 1) {
    ps += __shfl_xor(ps, off, 32);
    pd += __shfl_xor(pd, off, 32);
  }
  if (lane == 0) {
    s_src[node] = ps;
    s_dst[node] = pd;
  }
}

// ---------------------------------------------------------------------------
// Kernel 3: per-edge logit -> leaky relu -> clip -> exp; segment-sum the
// numerator into denom[dst] with a native f32 global atomic.
// ---------------------------------------------------------------------------
__global__ __launch_bounds__(256) void gat_edge_num_kernel(
    const long long* __restrict__ src, const long long* __restrict__ dst,
    const float* __restrict__ s_src, const float* __restrict__ s_dst,
    float* __restrict__ denom, float* __restrict__ alpha_num, int E) {
  const int e = blockIdx.x * blockDim.x + threadIdx.x;
  if (e >= E) return;
  const int s = (int)src[e];
  const int d = (int)dst[e];
  float v = s_src[s] + s_dst[d];
  v = (v >= 0.0f) ? v : 0.2f * v;             // LeakyReLU(0.2)
  v = fminf(fmaxf(v, -10.0f), 10.0f);         // clip
  v = __expf(v);
  alpha_num[e] = v;
  atomic_add_f32(&denom[d], v);
}

// ---------------------------------------------------------------------------
// Kernel 4: out[dst] += alpha * h[src].  One wave per edge; each lane owns
// columns lane and lane+32 (coalesced gather of h[src] and coalesced f32
// atomics into out[dst]).  h and out both fit in the 192MB L2.
// ---------------------------------------------------------------------------
__global__ __launch_bounds__(256) void gat_scatter_kernel(
    const long long* __restrict__ src, const long long* __restrict__ dst,
    const float* __restrict__ alpha_num, const float* __restrict__ denom,
    const float* __restrict__ h, float* __restrict__ out, int E) {
  const int e = (int)((blockIdx.x * (long)blockDim.x + threadIdx.x) >> 5);
  const int lane = threadIdx.x & 31;
  if (e >= E) return;
  const int s = (int)src[e];
  const int d = (int)dst[e];
  const float alpha = alpha_num[e] / (denom[d] + 1e-9f);
  const float* __restrict__ hs = h + (long)s * OUT_DIM;
  float* __restrict__ od = out + (long)d * OUT_DIM;
  atomic_add_f32(&od[lane], alpha * hs[lane]);
  atomic_add_f32(&od[lane + 32], alpha * hs[lane + 32]);
}

// ---------------------------------------------------------------------------
// Host launcher. Inputs (setup_inputs order):
//   d_in[0] x         f32 [N,128]
//   d_in[1] edge_index i64 [2,E]   (row 0 = src, row 1 = dst)
//   d_in[2] W         f32 [64,128]
//   d_in[3] a_src     f32 [64]
//   d_in[4] a_dst     f32 [64]
// Output: f32 [N,64].
// Workspace layout (floats): h[N*64] | s_src[N] | s_dst[N] | denom[N] | alpha[E]
// ---------------------------------------------------------------------------
extern "C" void kernel_launch(void* const* d_in, const int* in_sizes, int n_in,
                              void* d_out, int out_size, void* d_ws,
                              size_t ws_size, hipStream_t stream) {
  const float* x = (const float*)d_in[0];
  const long long* ei = (const long long*)d_in[1];
  const float* W = (const float*)d_in[2];
  const float* a_src = (const float*)d_in[3];
  const float* a_dst = (const float*)d_in[4];
  float* out = (float*)d_out;

  const int N = in_sizes[0] / IN_DIM;   // 100000
  const int E = in_sizes[1] / 2;        // 1600000
  const long long* src = ei;
  const long long* dst = ei + E;

  float* ws = (float*)d_ws;
  float* h = ws;
  float* s_src = h + (size_t)N * OUT_DIM;
  float* s_dst = s_src + N;
  float* denom = s_dst + N;
  float* alpha_num = denom + N;

  // Zero the atomic accumulation targets (graph-capture-safe memset nodes).
  hipMemsetAsync(denom, 0, (size_t)N * sizeof(float), stream);
  hipMemsetAsync(out, 0, (size_t)N * OUT_DIM * sizeof(float), stream);

  // 1) h = x @ W^T  (WMMA f32)
  {
    dim3 grid(N / 16);  // N divisible by 16 (100000 = 6250*16)
    gat_gemm_kernel<<<grid, 128, 0, stream>>>(x, W, h, N);
  }
  // 2) per-node logits
  {
    long waves = N;
    dim3 grid((unsigned)((waves * 32 + 255) / 256));
    gat_logits_kernel<<<grid, 256, 0, stream>>>(h, a_src, a_dst, s_src, s_dst,
                                                N);
  }
  // 3) edge numerators + denominator segment-sum
  {
    dim3 grid((E + 255) / 256);
    gat_edge_num_kernel<<<grid, 256, 0, stream>>>(src, dst, s_src, s_dst,
                                                  denom, alpha_num, E);
  }
  // 4) normalized scatter-add of messages
  {
    long waves = E;
    dim3 grid((unsigned)((waves * 32 + 255) / 256));
    gat_scatter_kernel<<<grid, 256, 0, stream>>>(src, dst, alpha_num, denom, h,
                                                 out, E);
  }
}